// CenterLoss_79096117723175
// MI455X (gfx1250) — compile-verified
//
#include <hip/hip_runtime.h>
#include <hip/hip_bf16.h>
#include <stdint.h>

#define NUM_CLASSES 100000
#define EMBED_DIM   512
#define BATCH       16384

// ---------------------------------------------------------------------------
// Kernel 1: bulk copy center_var -> new_centers (d_out+1) using the CDNA5
// async global<->LDS DMA path. Source is 16B aligned -> async b128 loads to
// LDS; destination is only 4B aligned (offset by the leading loss scalar) ->
// four async b32 stores from LDS. Non-temporal hints: this 400MB stream
// should not evict the L2 lines the gather kernel wants.
// Also zero-initializes the f64 loss accumulator in d_ws (thread 0).
// ---------------------------------------------------------------------------
__global__ __launch_bounds__(256) void copy_centers_async(
    const float* __restrict__ src, float* __restrict__ dst,
    double* __restrict__ loss_accum, long n4)
{
    __shared__ char stage[256 * 16];

    long gid = (long)blockIdx.x * blockDim.x + threadIdx.x;
    if (gid == 0) *loss_accum = 0.0;
    if (gid >= n4) return;

    // Low 32 bits of a generic pointer into __shared__ == LDS byte offset
    // (the shared aperture carries the offset in addr[31:0]).
    unsigned lds = (unsigned)(uintptr_t)(&stage[threadIdx.x * 16]);
    const char* s = (const char*)src + gid * 16;
    char*       d = (char*)dst + gid * 16;

    // 16B/lane global -> LDS, async (ASYNCcnt)
    asm volatile("global_load_async_to_lds_b128 %0, %1, off th:TH_LOAD_NT"
                 :: "v"(lds), "v"(s) : "memory");
    // LDS side of async ops can complete out of order -> wait before reading
    asm volatile("s_wait_asynccnt 0x0" ::: "memory");
    // 4 x 4B/lane LDS -> global, async; offset applies to both LDS and mem addr
    asm volatile("global_store_async_from_lds_b32 %0, %1, off th:TH_STORE_NT"
                 :: "v"(d), "v"(lds) : "memory");
    asm volatile("global_store_async_from_lds_b32 %0, %1, off offset:4 th:TH_STORE_NT"
                 :: "v"(d), "v"(lds) : "memory");
    asm volatile("global_store_async_from_lds_b32 %0, %1, off offset:8 th:TH_STORE_NT"
                 :: "v"(d), "v"(lds) : "memory");
    asm volatile("global_store_async_from_lds_b32 %0, %1, off offset:12 th:TH_STORE_NT"
                 :: "v"(d), "v"(lds) : "memory");
    // S_ENDPGM performs an implicit wait-idle, draining ASYNCcnt.
}

// ---------------------------------------------------------------------------
// Kernel 2: one 128-thread block per batch row (4 wave32 waves).
//   gather center row, compute diff = 0.05*(c - f),
//   atomically subtract diff from new_centers[label] (duplicate-label safe),
//   accumulate (f-c)^2 block partial -> one f64 atomic per block.
// ---------------------------------------------------------------------------
__global__ __launch_bounds__(128) void gather_scatter_loss(
    const float* __restrict__ features, const int* __restrict__ labels,
    const float* __restrict__ centers, float* __restrict__ out_centers,
    double* __restrict__ loss_accum)
{
    const int row = blockIdx.x;
    const int tid = threadIdx.x;
    const int label = labels[row];

    const float4* f4 = (const float4*)(features + (size_t)row * EMBED_DIM);
    const float4* c4 = (const float4*)(centers + (size_t)label * EMBED_DIM);
    float4 f = f4[tid];
    float4 c = c4[tid];

    float dx = c.x - f.x;
    float dy = c.y - f.y;
    float dz = c.z - f.z;
    float dw = c.w - f.w;

    const float kNegDelta = -(float)(1.0 - 0.95);  // -= (1-ALPHA)*(c-f)
    float* outp = out_centers + (size_t)label * EMBED_DIM + (size_t)tid * 4;
    __hip_atomic_fetch_add(outp + 0, kNegDelta * dx, __ATOMIC_RELAXED, __HIP_MEMORY_SCOPE_AGENT);
    __hip_atomic_fetch_add(outp + 1, kNegDelta * dy, __ATOMIC_RELAXED, __HIP_MEMORY_SCOPE_AGENT);
    __hip_atomic_fetch_add(outp + 2, kNegDelta * dz, __ATOMIC_RELAXED, __HIP_MEMORY_SCOPE_AGENT);
    __hip_atomic_fetch_add(outp + 3, kNegDelta * dw, __ATOMIC_RELAXED, __HIP_MEMORY_SCOPE_AGENT);

    // (f - c)^2 == (c - f)^2
    float ss = dx * dx + dy * dy + dz * dz + dw * dw;

    __shared__ float red[128];
    red[tid] = ss;
    __syncthreads();
#pragma unroll
    for (int off = 64; off > 0; off >>= 1) {
        if (tid < off) red[tid] += red[tid + off];
        __syncthreads();
    }
    if (tid == 0) {
        __hip_atomic_fetch_add(loss_accum, (double)red[0],
                               __ATOMIC_RELAXED, __HIP_MEMORY_SCOPE_AGENT);
    }
}

// ---------------------------------------------------------------------------
// Kernel 3: finalize the mean into d_out[0].
// ---------------------------------------------------------------------------
__global__ void finalize_loss(const double* __restrict__ loss_accum,
                              float* __restrict__ out)
{
    out[0] = (float)(*loss_accum * (1.0 / ((double)BATCH * (double)EMBED_DIM)));
}

extern "C" void kernel_launch(void* const* d_in, const int* in_sizes, int n_in,
                              void* d_out, int out_size, void* d_ws, size_t ws_size,
                              hipStream_t stream) {
    const float* features   = (const float*)d_in[0];
    const int*   labels     = (const int*)d_in[1];   // JAX canonicalizes int64->int32
    const float* center_var = (const float*)d_in[2];

    float* out         = (float*)d_out;   // [0] = loss, [1..] = new_centers
    float* out_centers = out + 1;
    double* loss_accum = (double*)d_ws;

    const long n4 = (long)NUM_CLASSES * EMBED_DIM / 4;   // 12,800,000 float4s
    const int  cpb = 256;
    const long cblocks = (n4 + cpb - 1) / cpb;           // 50,000 blocks

    copy_centers_async<<<(unsigned)cblocks, cpb, 0, stream>>>(
        center_var, out_centers, loss_accum, n4);
    gather_scatter_loss<<<BATCH, 128, 0, stream>>>(
        features, labels, center_var, out_centers, loss_accum);
    finalize_loss<<<1, 1, 0, stream>>>(loss_accum, out);
}